// CINConv_2688649527597
// MI455X (gfx1250) — compile-verified
//
#include <hip/hip_runtime.h>
#include <hip/hip_bf16.h>

typedef float v2f __attribute__((ext_vector_type(2)));
typedef float v8f __attribute__((ext_vector_type(8)));

// ---------------------------------------------------------------------------
// Native f32 global atomic add (global_atomic_add_f32, no CAS loop)
// ---------------------------------------------------------------------------
__device__ __forceinline__ void atomic_add_f32(float* p, float v) {
    __hip_atomic_fetch_add(p, v, __ATOMIC_RELAXED, __HIP_MEMORY_SCOPE_AGENT);
}

// ---------------------------------------------------------------------------
// Init: aggB = x, aggR = x, aggU = 0   (folds the (1+eps)*x identity term)
// ---------------------------------------------------------------------------
__global__ void init_agg_kernel(const float* __restrict__ x,
                                float* __restrict__ aggB,
                                float* __restrict__ aggR,
                                float* __restrict__ aggU,
                                long n128, long n256) {
    long i = (long)blockIdx.x * blockDim.x + threadIdx.x;
    if (i < n128) {
        float v = x[i];
        aggB[i] = v;
        aggR[i] = v;
    }
    if (i < n256) aggU[i] = 0.0f;
}

// ---------------------------------------------------------------------------
// Edge scatter: one wave (32 lanes) per edge; each lane handles 4 floats.
//   type 0 (BOUNDARY): aggB[dst] += x[src]
//   type 2 (REWIRE)  : aggR[dst] += x[src]
//   type 1 (UPPER)   : aggU[dst, 0:128] += x[src]; aggU[dst,128:256] += x[upper_ind]
// ---------------------------------------------------------------------------
__global__ void edge_scatter_kernel(const float* __restrict__ x,
                                    const int* __restrict__ ei,     // [2,E]
                                    const int* __restrict__ etype,  // [E]
                                    const int* __restrict__ uind,   // [E]
                                    float* __restrict__ aggB,
                                    float* __restrict__ aggR,
                                    float* __restrict__ aggU,
                                    int E_) {
    int gw   = (int)((blockIdx.x * (unsigned)blockDim.x + threadIdx.x) >> 5);
    int lane = threadIdx.x & 31;
    if (gw >= E_) return;

    int t   = etype[gw];
    int src = ei[gw];
    int dst = ei[E_ + gw];
    int c   = lane * 4;

    float4 v = *(const float4*)(x + (size_t)src * 128 + c);

    if (t == 1) {  // UPPER
        int u = uind[gw];
        float4 vu = *(const float4*)(x + (size_t)u * 128 + c);
        float* p = aggU + (size_t)dst * 256 + c;
        atomic_add_f32(p + 0, v.x);
        atomic_add_f32(p + 1, v.y);
        atomic_add_f32(p + 2, v.z);
        atomic_add_f32(p + 3, v.w);
        p += 128;
        atomic_add_f32(p + 0, vu.x);
        atomic_add_f32(p + 1, vu.y);
        atomic_add_f32(p + 2, vu.z);
        atomic_add_f32(p + 3, vu.w);
    } else {
        float* base = (t == 0) ? aggB : aggR;
        float* p = base + (size_t)dst * 128 + c;
        atomic_add_f32(p + 0, v.x);
        atomic_add_f32(p + 1, v.y);
        atomic_add_f32(p + 2, v.z);
        atomic_add_f32(p + 3, v.w);
    }
}

// ---------------------------------------------------------------------------
// GEMM: C[nrows,128] = act(A[nrows,K] @ W[K,128] + bias (+ residual[N,128]))
// 256 threads = 8 waves; block computes 128 rows x 128 cols.
// Wave w owns rows [rowBase + 16w, +16), all 8 column tiles of 16.
// Uses V_WMMA_F32_16X16X4_F32 (exact fp32 matrix math).
//
//   A layout: lane L -> row M = L&15, K-half koff = (L>=16)?2:0, 2 VGPRs {K,K+1}
//             -> one ds_load_b64 from sA (row-major, padded stride).
//   B layout: lane L -> col N = L&15, same K-half split, 2 VGPRs {K,K+1}
//             -> W staged K-PAIR INTERLEAVED: sW[kp*STRIDE + 2n + s] = W[2kp+s][n]
//                so {B[k][n], B[k+1][n]} is one aligned ds_load_b64 (no repack movs),
//                and the 8 column tiles are immediate offsets (nt*128 B) off one base.
//   C layout: lane L -> col N = L&15, row = base + (L>=16?8:0) + vgpr
// ---------------------------------------------------------------------------
#define SA_STRIDE  132   // dwords per A row (128 + pad)
#define SW_PSTRIDE 264   // dwords per K-pair row of W (2*128 + pad)

__launch_bounds__(256)
__global__ void gemm_wmma_kernel(const float* __restrict__ A, int lda, int K,
                                 const float* __restrict__ W,     // [K,128]
                                 const float* __restrict__ bias,  // [128]
                                 const float* __restrict__ residual, // [N,128] or null
                                 float* __restrict__ C, int ldc,
                                 int nrows, int do_relu) {
    extern __shared__ float lds[];
    float* sW = lds;                        // [64 k-pairs][SW_PSTRIDE]
    float* sA = lds + 64 * SW_PSTRIDE;      // [128 rows][SA_STRIDE]

    const int tid  = threadIdx.x;
    const int wave = tid >> 5;
    const int lane = tid & 31;
    const int rowBase = blockIdx.x * 128;

    v8f acc[8] = {};

    const int arow = wave * 16 + (lane & 15);
    const int koff = (lane >> 4) * 2;   // 0 or 2 : K offset within a 4-step
    const int khi  = (lane >> 4);       // 0 or 1 : K-pair-row offset
    const int ncol = lane & 15;

    for (int kc = 0; kc < K; kc += 128) {
        // stage W chunk [128 x 128] -> sW, K-pair interleaved
        for (int i = tid; i < 2048; i += 256) {
            int kp = i >> 5, n4 = (i & 31) << 2;
            const float* w0 = W + (size_t)(kc + 2 * kp) * 128 + n4;
            float4 r0 = *(const float4*)(w0);
            float4 r1 = *(const float4*)(w0 + 128);
            float* d = &sW[kp * SW_PSTRIDE + 2 * n4];
            v2f p0; p0.x = r0.x; p0.y = r1.x;
            v2f p1; p1.x = r0.y; p1.y = r1.y;
            v2f p2; p2.x = r0.z; p2.y = r1.z;
            v2f p3; p3.x = r0.w; p3.y = r1.w;
            ((v2f*)d)[0] = p0;
            ((v2f*)d)[1] = p1;
            ((v2f*)d)[2] = p2;
            ((v2f*)d)[3] = p3;
        }
        // stage A chunk [128 rows x 128 k] -> sA (padded, zero-fill OOB rows)
        for (int i = tid; i < 4096; i += 256) {
            int r = i >> 5, cc = (i & 31) << 2;
            int row = rowBase + r;
            float4 v = make_float4(0.f, 0.f, 0.f, 0.f);
            if (row < nrows)
                v = *(const float4*)(A + (size_t)row * lda + kc + cc);
            *(float4*)(&sA[r * SA_STRIDE + cc]) = v;
        }
        __syncthreads();

#pragma unroll 2
        for (int k = 0; k < 128; k += 4) {
            v2f a = *reinterpret_cast<const v2f*>(&sA[arow * SA_STRIDE + k + koff]);
            const float* bb = &sW[((k >> 1) + khi) * SW_PSTRIDE + 2 * ncol];
#pragma unroll
            for (int nt = 0; nt < 8; nt++) {
                v2f b = *reinterpret_cast<const v2f*>(bb + nt * 32);
                acc[nt] = __builtin_amdgcn_wmma_f32_16x16x4_f32(
                    false, a, false, b, (short)0, acc[nt], false, false);
            }
        }
        __syncthreads();
    }

    // epilogue: bias (+ residual) (+ relu), scatter per C/D VGPR layout
    const int rhalf = (lane >> 4) * 8;
#pragma unroll
    for (int nt = 0; nt < 8; nt++) {
        int col = nt * 16 + ncol;
        float bv = bias[col];
#pragma unroll
        for (int v = 0; v < 8; v++) {
            int row = rowBase + wave * 16 + rhalf + v;
            if (row < nrows) {
                float val = acc[nt][v] + bv;
                if (residual) val += residual[(size_t)row * 128 + col];
                if (do_relu) val = fmaxf(val, 0.0f);
                C[(size_t)row * ldc + col] = val;
            }
        }
    }
}

// ---------------------------------------------------------------------------
// Host launcher
// ---------------------------------------------------------------------------
extern "C" void kernel_launch(void* const* d_in, const int* in_sizes, int n_in,
                              void* d_out, int out_size, void* d_ws, size_t ws_size,
                              hipStream_t stream) {
    const float* x     = (const float*)d_in[0];
    const int*   ei    = (const int*)d_in[1];
    const int*   etype = (const int*)d_in[2];
    const int*   uind  = (const int*)d_in[3];
    // d_in[4] = cell_dimension (unused by reference)
    const float* bW1 = (const float*)d_in[5];
    const float* bb1 = (const float*)d_in[6];
    const float* bW2 = (const float*)d_in[7];
    const float* bb2 = (const float*)d_in[8];
    const float* rW1 = (const float*)d_in[9];
    const float* rb1 = (const float*)d_in[10];
    const float* rW2 = (const float*)d_in[11];
    const float* rb2 = (const float*)d_in[12];
    const float* umW = (const float*)d_in[13];
    const float* umb = (const float*)d_in[14];
    const float* uW1 = (const float*)d_in[15];
    const float* ub1 = (const float*)d_in[16];
    const float* uW2 = (const float*)d_in[17];
    const float* ub2 = (const float*)d_in[18];
    const float* oW  = (const float*)d_in[19];
    const float* ob  = (const float*)d_in[20];

    const int N_ = in_sizes[0] / 128;
    const int E_ = in_sizes[2];

    // workspace layout (floats): aggB[N,128] aggR[N,128] aggU[N,256]
    //                            h[N,128] upre[N,128] comb[N,384]
    float* ws   = (float*)d_ws;
    float* aggB = ws;
    float* aggR = aggB + (size_t)N_ * 128;
    float* aggU = aggR + (size_t)N_ * 128;
    float* hbuf = aggU + (size_t)N_ * 256;
    float* upre = hbuf + (size_t)N_ * 128;
    float* comb = upre + (size_t)N_ * 128;
    float* out  = (float*)d_out;

    // 1) init accumulators (identity term folded in)
    {
        long total = (long)N_ * 256;
        int blocks = (int)((total + 255) / 256);
        init_agg_kernel<<<blocks, 256, 0, stream>>>(x, aggB, aggR, aggU,
                                                    (long)N_ * 128, total);
    }

    // 2) fused edge scatter (one wave per edge)
    {
        long threads = (long)E_ * 32;
        int blocks = (int)((threads + 255) / 256);
        edge_scatter_kernel<<<blocks, 256, 0, stream>>>(x, ei, etype, uind,
                                                        aggB, aggR, aggU, E_);
    }

    // 3) GEMM chain (all via fp32 WMMA)
    const dim3 g((N_ + 127) / 128);
    const size_t shmem =
        (size_t)(64 * SW_PSTRIDE + 128 * SA_STRIDE) * sizeof(float);

    auto gemm = [&](const float* A, int lda, int K, const float* W,
                    const float* b, const float* res, float* Cp, int ldc, int relu) {
        gemm_wmma_kernel<<<g, 256, shmem, stream>>>(A, lda, K, W, b, res,
                                                    Cp, ldc, N_, relu);
    };

    // boundary branch -> comb[:, 0:128]
    gemm(aggB, 128, 128, bW1, bb1, nullptr, hbuf, 128, 1);
    gemm(hbuf, 128, 128, bW2, bb2, nullptr, comb + 0, 384, 0);
    // rewire branch -> comb[:, 128:256]
    gemm(aggR, 128, 128, rW1, rb1, nullptr, hbuf, 128, 1);
    gemm(hbuf, 128, 128, rW2, rb2, nullptr, comb + 128, 384, 0);
    // upper branch: msg linear (K=256, +x residual), then MLP -> comb[:, 256:384]
    gemm(aggU, 256, 256, umW, umb, x, upre, 128, 0);
    gemm(upre, 128, 128, uW1, ub1, nullptr, hbuf, 128, 1);
    gemm(hbuf, 128, 128, uW2, ub2, nullptr, comb + 256, 384, 0);
    // output layer: relu(comb[N,384] @ oW[384,128] + ob)
    gemm(comb, 384, 384, oW, ob, nullptr, out, 128, 1);
}